// PCNN_13099650253313
// MI455X (gfx1250) — compile-verified
//
#include <hip/hip_runtime.h>
#include <math.h>

typedef __attribute__((ext_vector_type(2))) float v2f;
typedef __attribute__((ext_vector_type(8))) float v8f;

#define LSEQ    128
#define CIN     150
#define HID     230
#define KDIM    450      // 3 * CIN, kk = k*150 + c (segment-major)
#define KPAD    452      // padded to multiple of 4
#define NPAD    256      // padded to 16 N-tiles (wave w owns tiles 2w, 2w+1)
#define XROWS   130      // 128 + 2 halo rows (pad=1 conv)
#define XSTRIDE 156      // 150 data cols + 6 zero cols; ≡4 mod 8 -> b64 bank-free
#define WS_ELEMS ((KPAD/2) * NPAD * 2)   // Wt2[kp][h][2]

// ---------------------------------------------------------------------------
// Repack W [H=230][C=150][3] -> Wt2 [KPAD/2=226][NPAD=256][2]:
//   Wt2[kp][h][v] = W[h][c][k]  with kk = 2*kp+v = k*150 + c,  zero padded.
// A lane's two B values (K0, K0+1) become one coalesced b64 load.
// ---------------------------------------------------------------------------
__global__ __launch_bounds__(256) void repack_w_kernel(
    const float* __restrict__ W, float* __restrict__ Wt2)
{
    int idx = blockIdx.x * blockDim.x + threadIdx.x;
    if (idx >= WS_ELEMS) return;
    int kp  = idx >> 9;           // / (NPAD*2)
    int rem = idx & 511;
    int h   = rem >> 1;
    int v   = rem & 1;
    int kk  = kp * 2 + v;
    float val = 0.0f;
    if (kk < KDIM && h < HID) {
        int k = kk / 150;
        int c = kk - k * 150;
        val = W[h * KDIM + c * 3 + k];
    }
    Wt2[idx] = val;
}

// ---------------------------------------------------------------------------
// One workgroup per batch element. Each wave computes TWO 16-wide N-tiles
// across all eight 16-row M-tiles (M=128, K=452pad) with f32 WMMA, then the
// piecewise masked max-pool + tanh straight from the accumulators.
// ---------------------------------------------------------------------------
__global__ __launch_bounds__(256) void pcnn_kernel(
    const float* __restrict__ Xea,    // [B][L][C]
    const int*   __restrict__ Xmask,  // [B][L]
    const float* __restrict__ Wt2,    // [KPAD/2][NPAD][2]
    const float* __restrict__ bias,   // [H]
    float*       __restrict__ out)    // [B][3*H]
{
    extern __shared__ char smem_raw[];
    float* Xls    = (float*)smem_raw;                       // [XROWS][XSTRIDE]
    int*   maskSh = (int*)(smem_raw + XROWS * XSTRIDE * 4); // [LSEQ]

    const int b      = blockIdx.x;
    const int tid    = threadIdx.x;
    const int lane   = tid & 31;
    const int wave   = tid >> 5;     // 8 waves
    const int nlane  = lane & 15;    // N index (B/C/D) and M index (A)
    const int laneHi = lane >> 4;    // selects K pair

    // ---- zero LDS (halo rows + pad cols 150..155 must be 0) ----
    for (int i = tid; i < XROWS * XSTRIDE; i += 256) Xls[i] = 0.0f;
    __syncthreads();

    // ---- stage X[b] into LDS rows 1..128, cols 0..149 ----
    const float* Xb = Xea + (size_t)b * (LSEQ * CIN);
    for (int i = tid; i < LSEQ * CIN; i += 256) {
        int l = i / CIN;
        int c = i - l * CIN;
        Xls[(l + 1) * XSTRIDE + c] = Xb[i];
    }
    if (tid < LSEQ) maskSh[tid] = Xmask[(size_t)b * LSEQ + tid];
    __syncthreads();

    // ---- this wave's two N-tiles ----
    const int   jt0 = wave * 2, jt1 = wave * 2 + 1;   // jt1 == 15 is zero pad
    const int   h0  = jt0 * 16 + nlane;
    const int   h1  = jt1 * 16 + nlane;
    const float bh0 = (h0 < HID) ? bias[h0] : 0.0f;
    const float bh1 = (h1 < HID) ? bias[h1] : 0.0f;

    v8f acc0[8], acc1[8];
    #pragma unroll
    for (int mt = 0; mt < 8; ++mt) {
        #pragma unroll
        for (int r = 0; r < 8; ++r) { acc0[mt][r] = bh0; acc1[mt][r] = bh1; }
    }

    const float* arowBase = Xls + nlane * XSTRIDE;
    const int    bBase    = laneHi * (NPAD * 2) + nlane * 2;   // + (ks/2)*512 + jt*32

    // ---- K loop: 113 steps of K=4, 16 WMMAs per step ----
    #pragma unroll 2
    for (int ks = 0; ks < KPAD; ks += 4) {
        const int K0 = ks + 2 * laneHi;                 // lane's even K start
        const int k  = (K0 >= 150) + (K0 >= 300);       // conv tap (tail -> c=150 zeros)
        const int aoff = k * XSTRIDE + (K0 - 150 * k);  // same row, consecutive cols
        const float* arow = arowBase + aoff;

        v2f af[8];                                      // distinct regs -> one wait
        #pragma unroll
        for (int mt = 0; mt < 8; ++mt)
            af[mt] = *(const v2f*)(arow + mt * 16 * XSTRIDE);   // ds_load_b64

        const float* bptr = Wt2 + (size_t)(ks >> 1) * (NPAD * 2) + bBase;
        v2f bf0 = *(const v2f*)(bptr + jt0 * 32);               // global b64 (L2)
        v2f bf1 = *(const v2f*)(bptr + jt1 * 32);

        #pragma unroll
        for (int mt = 0; mt < 8; ++mt) {
            acc0[mt] = __builtin_amdgcn_wmma_f32_16x16x4_f32(
                false, af[mt], false, bf0, (short)0, acc0[mt], false, false);
            acc1[mt] = __builtin_amdgcn_wmma_f32_16x16x4_f32(
                false, af[mt], false, bf1, (short)0, acc1[mt], false, false);
        }
    }

    // ---- piecewise masked max-pool over l, from registers ----
    float p00 = -INFINITY, p01 = -INFINITY, p02 = -INFINITY;
    float p10 = -INFINITY, p11 = -INFINITY, p12 = -INFINITY;
    #pragma unroll
    for (int mt = 0; mt < 8; ++mt) {
        #pragma unroll
        for (int r = 0; r < 8; ++r) {
            const int   l  = mt * 16 + r + 8 * laneHi;  // C/D: M = r + 8*laneHi
            const int   mv = maskSh[l];
            const float v0 = acc0[mt][r];
            const float v1 = acc1[mt][r];
            p00 = fmaxf(p00, (mv == 1) ? v0 : 0.0f);
            p01 = fmaxf(p01, (mv == 2) ? v0 : 0.0f);
            p02 = fmaxf(p02, (mv == 3) ? v0 : 0.0f);
            p10 = fmaxf(p10, (mv == 1) ? v1 : 0.0f);
            p11 = fmaxf(p11, (mv == 2) ? v1 : 0.0f);
            p12 = fmaxf(p12, (mv == 3) ? v1 : 0.0f);
        }
    }
    // merge the two lane-halves (same h, complementary l sets)
    p00 = fmaxf(p00, __shfl_xor(p00, 16, 32));
    p01 = fmaxf(p01, __shfl_xor(p01, 16, 32));
    p02 = fmaxf(p02, __shfl_xor(p02, 16, 32));
    p10 = fmaxf(p10, __shfl_xor(p10, 16, 32));
    p11 = fmaxf(p11, __shfl_xor(p11, 16, 32));
    p12 = fmaxf(p12, __shfl_xor(p12, 16, 32));

    if (laneHi == 0) {
        float* ob = out + (size_t)b * (3 * HID);
        if (h0 < HID) {
            float* o = ob + h0 * 3;
            o[0] = tanhf(p00); o[1] = tanhf(p01); o[2] = tanhf(p02);
        }
        if (h1 < HID) {
            float* o = ob + h1 * 3;
            o[0] = tanhf(p10); o[1] = tanhf(p11); o[2] = tanhf(p12);
        }
    }
}

// ---------------------------------------------------------------------------
extern "C" void kernel_launch(void* const* d_in, const int* in_sizes, int n_in,
                              void* d_out, int out_size, void* d_ws, size_t ws_size,
                              hipStream_t stream)
{
    const float* Xea   = (const float*)d_in[0];
    const int*   Xmask = (const int*)  d_in[1];
    const float* W     = (const float*)d_in[2];
    const float* bias  = (const float*)d_in[3];
    float*       out   = (float*)d_out;
    float*       Wt2   = (float*)d_ws;   // WS_ELEMS floats = 452 KB

    const int B = in_sizes[0] / (LSEQ * CIN);   // 2048

    repack_w_kernel<<<(WS_ELEMS + 255) / 256, 256, 0, stream>>>(W, Wt2);

    const size_t smem = (size_t)XROWS * XSTRIDE * sizeof(float)
                      + (size_t)LSEQ * sizeof(int);   // ~81.6 KB
    pcnn_kernel<<<B, 256, smem, stream>>>(Xea, Xmask, Wt2, bias, out);
}